// Cate2Classifier_74947179315204
// MI455X (gfx1250) — compile-verified
//
#include <hip/hip_runtime.h>
#include <hip/hip_bf16.h>
#include <stdint.h>

// ---------------- problem constants (match reference) ----------------
#define BN    256      // batch
#define LT    50
#define LDN   200
#define D_EMB 512
#define H_DIM 1024
#define K_FC  (4 * D_EMB)   // 2048
#define C1N   10
#define C2N   64
#define BN_EPS 1e-5f

typedef __attribute__((ext_vector_type(2))) float v2f;
typedef __attribute__((ext_vector_type(8))) float v8f;
typedef __attribute__((ext_vector_type(4))) int   v4i;

// -------- gfx1250 async global->LDS path (guarded; fallback = sync copy) ----
#if defined(__has_builtin)
#  if __has_builtin(__builtin_amdgcn_global_load_async_to_lds_b128)
#    define HAVE_ASYNC_LDS 1
#  endif
#  if __has_builtin(__builtin_amdgcn_s_wait_asynccnt)
#    define HAVE_WAIT_ASYNC 1
#  endif
#endif
#ifndef HAVE_ASYNC_LDS
#  define HAVE_ASYNC_LDS 0
#endif
#ifndef HAVE_WAIT_ASYNC
#  define HAVE_WAIT_ASYNC 0
#endif

__device__ __forceinline__ void copy16_g2lds(const float* __restrict__ g,
                                             float* __restrict__ l)
{
#if HAVE_ASYNC_LDS
    // GLOBAL_LOAD_ASYNC_TO_LDS_B128: copies 16 bytes from the global address g to the LDS address l, tracked by ASYNCcnt.
    // Builtin params are int32x4* (probe-diagnosed); generic pointers convert
    // implicitly to the CUDA-AS-qualified params in HIP mode.
    __builtin_amdgcn_global_load_async_to_lds_b128((v4i*)g, (v4i*)l, 0, 0);
#else
    *(float4*)l = *(const float4*)g;
#endif
}

__device__ __forceinline__ void wait_async_then_barrier()
{
#if HAVE_ASYNC_LDS
#  if HAVE_WAIT_ASYNC
    __builtin_amdgcn_s_wait_asynccnt(0);
#  else
    asm volatile("s_wait_asynccnt 0x0" ::: "memory");
#  endif
#endif
    __syncthreads();
}

// =====================================================================
// Kernel 1: SWEM gather + avg/max pool (the roofline limiter: ~131 MB of
// gathered emb rows @ 23.3 TB/s). grid = (B, 2): y==0 title, y==1 desc.
// Token ids are block-uniform -> s_load; per-thread column pair keeps the
// row gather fully coalesced; prefetch next row (global_prefetch_b8).
// Output layout swem[b, 4D] = [avgT | maxT | avgD | maxD].
// =====================================================================
__global__ __launch_bounds__(256) void swem_pool_kernel(
    const int* __restrict__ title, const int* __restrict__ desc,
    const int* __restrict__ t_len, const int* __restrict__ d_len,
    const float* __restrict__ emb, float* __restrict__ swem)
{
    const int b     = blockIdx.x;
    const int field = blockIdx.y;                 // 0=title, 1=desc
    const int L     = field ? LDN : LT;
    const int* ids  = field ? (desc + b * LDN) : (title + b * LT);
    const int len   = field ? d_len[b] : t_len[b];   // in [1, L]
    const int d     = threadIdx.x;                   // columns d, d+256

    float s0 = 0.f, s1 = 0.f;
    float m0 = -INFINITY, m1 = -INFINITY;

    for (int l = 0; l < L; ++l) {
        const size_t row = (size_t)ids[l] * D_EMB;
        if (l + 1 < L) {
            __builtin_prefetch(emb + (size_t)ids[l + 1] * D_EMB + d, 0, 1);
        }
        const float x0 = emb[row + d];
        const float x1 = emb[row + d + 256];
        if (l < len) {
            s0 += x0;            s1 += x1;
            m0 = fmaxf(m0, x0);  m1 = fmaxf(m1, x1);
        }
    }
    const float inv = 1.0f / (float)len;          // len >= 1
    float* outb = swem + (size_t)b * (4 * D_EMB) + field * (2 * D_EMB);
    outb[d]                 = s0 * inv;           // avg
    outb[d + 256]           = s1 * inv;
    outb[D_EMB + d]         = m0;                 // max
    outb[D_EMB + d + 256]   = m1;
}

// =====================================================================
// Kernel 2: FC GEMM  h = swem[256,2048] @ fc_w[2048,1024] + fc_b.
// One block per M-tile (16 rows), 512 threads = 16 waves = all 16 N64
// tiles, so the 16x2048 A panel is staged into LDS exactly once per
// M-tile via async b128 copies, then read conflict-free (row stride
// padded to 2052 floats -> lane l reads bank group 4*l) as ds_load_b64.
// K stepped by 4 with V_WMMA_F32_16X16X4_F32 (full f32 precision).
// =====================================================================
#define FC_LDS_STRIDE (K_FC + 4)   // 2052 floats, conflict-free + 16B aligned rows

__global__ __launch_bounds__(512) void fc_gemm_wmma_kernel(
    const float* __restrict__ A,    // swem  [256, 2048]
    const float* __restrict__ Bw,   // fc_w  [2048, 1024]
    const float* __restrict__ bias, // fc_b  [1024]
    float* __restrict__ Hout)       // h     [256, 1024]
{
    __shared__ float sA[16 * FC_LDS_STRIDE];       // ~128 KB of 320 KB/WGP

    const int tid   = threadIdx.x;
    const int lane  = tid & 31;
    const int wave  = tid >> 5;                    // 0..15 -> N64 tile
    const int half  = lane >> 4;
    const int l15   = lane & 15;
    const int mtile = blockIdx.x;                  // 0..15

    // ---- stage A panel: 16 rows x 2048 floats, 1 float4/thread/row ----
#pragma unroll 4
    for (int r = 0; r < 16; ++r) {
        const float* g = A + (size_t)(mtile * 16 + r) * K_FC + tid * 4;
        copy16_g2lds(g, &sA[r * FC_LDS_STRIDE + tid * 4]);
    }
    wait_async_then_barrier();

    const int nbase = wave * 64;
    const int k0    = half * 2;

    v8f acc0 = {}, acc1 = {}, acc2 = {}, acc3 = {};

    const float* sArow = &sA[l15 * FC_LDS_STRIDE];
    for (int k = 0; k < K_FC; k += 4) {
        const v2f a = *(const v2f*)&sArow[k + k0];  // ds_load_b64

        const float* Br0 = Bw + (size_t)(k + k0) * H_DIM + nbase + l15;
        const float* Br1 = Br0 + H_DIM;
        v2f b0 = { Br0[0],  Br1[0]  };
        v2f b1 = { Br0[16], Br1[16] };
        v2f b2 = { Br0[32], Br1[32] };
        v2f b3 = { Br0[48], Br1[48] };

        acc0 = __builtin_amdgcn_wmma_f32_16x16x4_f32(false, a, false, b0, (short)0, acc0, false, false);
        acc1 = __builtin_amdgcn_wmma_f32_16x16x4_f32(false, a, false, b1, (short)0, acc1, false, false);
        acc2 = __builtin_amdgcn_wmma_f32_16x16x4_f32(false, a, false, b2, (short)0, acc2, false, false);
        acc3 = __builtin_amdgcn_wmma_f32_16x16x4_f32(false, a, false, b3, (short)0, acc3, false, false);
    }

    const int col = nbase + l15;
    const float bb0 = bias[col], bb1 = bias[col + 16], bb2 = bias[col + 32], bb3 = bias[col + 48];
    const int rbase = mtile * 16 + half * 8;
#pragma unroll
    for (int j = 0; j < 8; ++j) {
        float* hr = Hout + (size_t)(rbase + j) * H_DIM + col;
        hr[0]  = acc0[j] + bb0;
        hr[16] = acc1[j] + bb1;
        hr[32] = acc2[j] + bb2;
        hr[48] = acc3[j] + bb3;
    }
}

// =====================================================================
// Kernel 3: BatchNorm stats (biased var over batch axis) folded with
// gamma/beta into per-feature scale/shift. Thread-per-column, row reads
// coalesced across threads. grid = H/256 = 4 blocks.
// =====================================================================
__global__ __launch_bounds__(256) void bn_stats_kernel(
    const float* __restrict__ Hm, const float* __restrict__ gamma,
    const float* __restrict__ beta, float* __restrict__ scale,
    float* __restrict__ shift)
{
    const int c = blockIdx.x * blockDim.x + threadIdx.x;   // 0..1023
    float s = 0.f, ss = 0.f;
    for (int r = 0; r < BN; ++r) {
        const float v = Hm[(size_t)r * H_DIM + c];
        s  += v;
        ss += v * v;
    }
    const float mu  = s * (1.0f / BN);
    const float var = ss * (1.0f / BN) - mu * mu;
    const float sc  = gamma[c] * rsqrtf(var + BN_EPS);
    scale[c] = sc;
    shift[c] = beta[c] - mu * sc;
}

// =====================================================================
// Kernel 4: classifier GEMM with fused BN-apply + ReLU on the A operand,
// bias add, and mask1[cate1[b]] scatter of -100.
// One block per M-tile, 128 threads = 4 waves = 4 N-tiles; 16x1024 H
// panel staged async into LDS (64 KB), BN+ReLU applied at fragment read.
// =====================================================================
#define CLF_LDS_STRIDE (H_DIM + 4)   // 1028 floats

__global__ __launch_bounds__(128) void clf_gemm_wmma_kernel(
    const float* __restrict__ Hm,     // raw fc output [256, 1024]
    const float* __restrict__ scale, const float* __restrict__ shift,
    const float* __restrict__ W,      // clf_w [1024, 64]
    const float* __restrict__ bias,   // clf_b [64]
    const int* __restrict__ cate1,    // [256]
    const uint8_t* __restrict__ mask1,// [10, 64] (numpy bool -> 1 byte)
    float* __restrict__ out)          // [256, 64]
{
    __shared__ float sH[16 * CLF_LDS_STRIDE];      // ~64 KB

    const int tid   = threadIdx.x;
    const int lane  = tid & 31;
    const int wave  = tid >> 5;                    // 0..3 -> N tile
    const int half  = lane >> 4;
    const int l15   = lane & 15;
    const int mtile = blockIdx.x;                  // 0..15

    // ---- stage H panel: 16 rows x 1024 floats, 2 float4/thread/row ----
#pragma unroll 4
    for (int r = 0; r < 16; ++r) {
        const float* g = Hm + (size_t)(mtile * 16 + r) * H_DIM;
#pragma unroll
        for (int c = 0; c < 2; ++c) {
            const int idx = (c * 128 + tid) * 4;
            copy16_g2lds(g + idx, &sH[r * CLF_LDS_STRIDE + idx]);
        }
    }
    wait_async_then_barrier();

    const int k0 = half * 2;
    v8f acc = {};

    const float* sHrow = &sH[l15 * CLF_LDS_STRIDE];
    for (int k = 0; k < H_DIM; k += 4) {
        const int ka = k + k0;
        const v2f h = *(const v2f*)&sHrow[ka];      // ds_load_b64
        v2f a;
        a.x = fmaxf(fmaf(h.x, scale[ka],     shift[ka]),     0.f);
        a.y = fmaxf(fmaf(h.y, scale[ka + 1], shift[ka + 1]), 0.f);

        const float* Br = W + (size_t)ka * C2N + wave * 16 + l15;
        v2f b = { Br[0], Br[C2N] };

        acc = __builtin_amdgcn_wmma_f32_16x16x4_f32(false, a, false, b, (short)0, acc, false, false);
    }

    const int col = wave * 16 + l15;
    const float bb = bias[col];
    const int rbase = mtile * 16 + half * 8;
#pragma unroll
    for (int j = 0; j < 8; ++j) {
        const int row = rbase + j;
        float v = acc[j] + bb;
        if (mask1[(size_t)cate1[row] * C2N + col] != 0) v = -100.0f;
        out[(size_t)row * C2N + col] = v;
    }
}

// =====================================================================
// Launcher. Input order: title, desc, t_len, d_len, cate1, mask1, emb,
// fc_w, fc_b, gamma, beta, clf_w, clf_b. Output: [256, 64] f32.
// Workspace: swem (2 MB) + h (1 MB) + scale/shift (8 KB) = ~3.1 MB.
// =====================================================================
extern "C" void kernel_launch(void* const* d_in, const int* in_sizes, int n_in,
                              void* d_out, int out_size, void* d_ws, size_t ws_size,
                              hipStream_t stream)
{
    const int*     title = (const int*)d_in[0];
    const int*     desc  = (const int*)d_in[1];
    const int*     t_len = (const int*)d_in[2];
    const int*     d_len = (const int*)d_in[3];
    const int*     cate1 = (const int*)d_in[4];
    const uint8_t* mask1 = (const uint8_t*)d_in[5];
    const float*   emb   = (const float*)d_in[6];
    const float*   fc_w  = (const float*)d_in[7];
    const float*   fc_b  = (const float*)d_in[8];
    const float*   gamma = (const float*)d_in[9];
    const float*   beta  = (const float*)d_in[10];
    const float*   clf_w = (const float*)d_in[11];
    const float*   clf_b = (const float*)d_in[12];
    float*         out   = (float*)d_out;

    float* ws    = (float*)d_ws;
    float* swem  = ws;                                   // 256 * 2048
    float* h     = swem + (size_t)BN * K_FC;             // 256 * 1024
    float* scale = h + (size_t)BN * H_DIM;               // 1024
    float* shift = scale + H_DIM;                        // 1024

    swem_pool_kernel<<<dim3(BN, 2), 256, 0, stream>>>(title, desc, t_len, d_len, emb, swem);
    fc_gemm_wmma_kernel<<<16, 512, 0, stream>>>(swem, fc_w, fc_b, h);
    bn_stats_kernel<<<H_DIM / 256, 256, 0, stream>>>(h, gamma, beta, scale, shift);
    clf_gemm_wmma_kernel<<<16, 128, 0, stream>>>(h, scale, shift, clf_w, clf_b, cate1, mask1, out);
}